// QFCModelGen100_65481071404417
// MI455X (gfx1250) — compile-verified
//
#include <hip/hip_runtime.h>
#include <hip/hip_bf16.h>
#include <math.h>

typedef float v2f __attribute__((ext_vector_type(2)));
typedef float v8f __attribute__((ext_vector_type(8)));

#define HW 224
#define NPIX (224 * 224)

// ---------------------------------------------------------------------------
// Stage 1: fused conv(1->4,3x3,SAME) + relu + global average pool.
// One block per image, 224 threads (7 wave32s), one output column per thread.
// Register-sliding 3-row window, y unrolled by 3 so the window rotates through
// register roles with no shifting, no LDS staging, no in-loop barriers.
// ---------------------------------------------------------------------------
__global__ __launch_bounds__(224) void conv_pool_kernel(
    const float* __restrict__ x, const float* __restrict__ w,
    float* __restrict__ feat) {
  __shared__ float ssum[4];
  const int b = blockIdx.x;
  const int t = threadIdx.x;  // output column, 0..223
  const float* img = x + (size_t)b * NPIX;

  // weights (uniform across block -> scalar loads)
  float W[4][9];
#pragma unroll
  for (int c = 0; c < 4; c++)
#pragma unroll
    for (int k = 0; k < 9; k++) W[c][k] = w[c * 9 + k];

  if (t < 4) ssum[t] = 0.0f;

  float acc0 = 0.f, acc1 = 0.f, acc2 = 0.f, acc3 = 0.f;

  // one row-triple = (left, center, right) taps for this thread's column
  float al, ac, ar;  // row slot A
  float bl, bc, br;  // row slot B
  float cl, cc, cr;  // row slot C

#define LOAD_ROW(r, L, C, R)                                  \
  do {                                                        \
    const float* p = img + (r) * HW + t;                      \
    L = (t > 0) ? p[-1] : 0.0f;                               \
    C = p[0];                                                 \
    R = (t < HW - 1) ? p[1] : 0.0f;                           \
  } while (0)

#define ACCUM(v0, v1, v2, v3, v4, v5, v6, v7, v8)                            \
  do {                                                                       \
    float c0 = W[0][0] * v0 + W[0][1] * v1 + W[0][2] * v2 + W[0][3] * v3 +   \
               W[0][4] * v4 + W[0][5] * v5 + W[0][6] * v6 + W[0][7] * v7 +   \
               W[0][8] * v8;                                                 \
    float c1 = W[1][0] * v0 + W[1][1] * v1 + W[1][2] * v2 + W[1][3] * v3 +   \
               W[1][4] * v4 + W[1][5] * v5 + W[1][6] * v6 + W[1][7] * v7 +   \
               W[1][8] * v8;                                                 \
    float c2 = W[2][0] * v0 + W[2][1] * v1 + W[2][2] * v2 + W[2][3] * v3 +   \
               W[2][4] * v4 + W[2][5] * v5 + W[2][6] * v6 + W[2][7] * v7 +   \
               W[2][8] * v8;                                                 \
    float c3 = W[3][0] * v0 + W[3][1] * v1 + W[3][2] * v2 + W[3][3] * v3 +   \
               W[3][4] * v4 + W[3][5] * v5 + W[3][6] * v6 + W[3][7] * v7 +   \
               W[3][8] * v8;                                                 \
    acc0 += fmaxf(c0, 0.0f);                                                 \
    acc1 += fmaxf(c1, 0.0f);                                                 \
    acc2 += fmaxf(c2, 0.0f);                                                 \
    acc3 += fmaxf(c3, 0.0f);                                                 \
  } while (0)

  // invariant entering iteration i: A = row 3i-1, B = row 3i
  al = ac = ar = 0.0f;       // row -1 (top padding)
  LOAD_ROW(0, bl, bc, br);   // row 0

  for (int i = 0; i < 74; i++) {  // handles y = 3i, 3i+1, 3i+2 (0..221)
    const int y = 3 * i;
    if (t < 28 && y + 4 < HW)  // pull a future row toward the caches
      __builtin_prefetch(img + (y + 4) * HW + t * 8, 0, 1);
    LOAD_ROW(y + 1, cl, cc, cr);
    ACCUM(al, ac, ar, bl, bc, br, cl, cc, cr);  // pixel row y
    LOAD_ROW(y + 2, al, ac, ar);
    ACCUM(bl, bc, br, cl, cc, cr, al, ac, ar);  // pixel row y+1
    LOAD_ROW(y + 3, bl, bc, br);
    ACCUM(cl, cc, cr, al, ac, ar, bl, bc, br);  // pixel row y+2
  }
  // after loop: A = row 221, B = row 222
  LOAD_ROW(223, cl, cc, cr);
  ACCUM(al, ac, ar, bl, bc, br, cl, cc, cr);           // pixel row 222
  ACCUM(bl, bc, br, cl, cc, cr, 0.f, 0.f, 0.f);        // pixel row 223 (bottom pad)

#undef LOAD_ROW
#undef ACCUM

  __syncthreads();
  atomicAdd(&ssum[0], acc0);
  atomicAdd(&ssum[1], acc1);
  atomicAdd(&ssum[2], acc2);
  atomicAdd(&ssum[3], acc3);
  __syncthreads();
  if (t < 4) feat[b * 4 + t] = ssum[t] * (1.0f / (float)NPIX);
}

// ---------------------------------------------------------------------------
// Stage 2: build the fixed 16x16 layer unitary U = U_r + i*U_i from params.
// Thread j simulates the circuit on basis state e_j (column j of U).
// Wire w <-> bit (3-w) (reference reshape puts wire 0 on the leading axis).
// ---------------------------------------------------------------------------
__global__ void build_unitary_kernel(const float* __restrict__ params,
                                     float* __restrict__ Ur,
                                     float* __restrict__ Ui) {
  const int j = threadIdx.x;
  if (j >= 16) return;
  float re[16], im[16];
#pragma unroll
  for (int i = 0; i < 16; i++) { re[i] = 0.f; im[i] = 0.f; }
  re[j] = 1.f;
#pragma unroll
  for (int l = 0; l < 3; l++) {
#pragma unroll
    for (int w = 0; w < 4; w++) {
      const int bit = 1 << (3 - w);
      const float th = params[(l * 4 + w) * 2 + 0] * 0.5f;
      const float c = cosf(th), s = sinf(th);
#pragma unroll
      for (int s0 = 0; s0 < 16; s0++) {
        if (s0 & bit) continue;
        const int s1 = s0 | bit;
        const float ar = re[s0], ai = im[s0], br = re[s1], bi = im[s1];
        re[s0] = c * ar - s * br; im[s0] = c * ai - s * bi;
        re[s1] = s * ar + c * br; im[s1] = s * ai + c * bi;
      }
      const float ph = params[(l * 4 + w) * 2 + 1] * 0.5f;
      const float cz = cosf(ph), sz = sinf(ph);
#pragma unroll
      for (int s0 = 0; s0 < 16; s0++) {
        const float pi = (s0 & bit) ? sz : -sz;  // e^{-i*ph} on bit=0
        const float ar = re[s0], ai = im[s0];
        re[s0] = ar * cz - ai * pi;
        im[s0] = ar * pi + ai * cz;
      }
    }
#pragma unroll
    for (int w = 0; w < 3; w++) {  // CNOT(w, w+1)
      const int cbit = 1 << (3 - w), tbit = 1 << (2 - w);
#pragma unroll
      for (int s0 = 0; s0 < 16; s0++) {
        if ((s0 & cbit) && !(s0 & tbit)) {
          const int s1 = s0 | tbit;
          const float tr = re[s0], ti = im[s0];
          re[s0] = re[s1]; im[s0] = im[s1];
          re[s1] = tr;     im[s1] = ti;
        }
      }
    }
  }
#pragma unroll
  for (int i = 0; i < 16; i++) {
    Ur[i * 16 + j] = re[i];
    Ui[i * 16 + j] = im[i];
  }
}

// ---------------------------------------------------------------------------
// Stage 3: batched amplitudes via WMMA.  Amp = (U_r + i U_i) * Psi, where
// Psi(16 x 512) is the real encoded product state.  One wave per 16-batch
// tile; 4 K-chunks x {real,imag} = 8 x v_wmma_f32_16x16x4_f32 per wave.
// A(16x4) lane layout: lanes 0-15 row=lane, K=k0,k0+1; lanes 16-31 K=k0+2,+3.
// B(4x16): lane = column, same K split.  D: vgpr v -> state M=v+8*half.
// ---------------------------------------------------------------------------
__global__ __launch_bounds__(32) void qexp_wmma_kernel(
    const float* __restrict__ feat, const float* __restrict__ Ur,
    const float* __restrict__ Ui, float* __restrict__ raw) {
  const int lane = threadIdx.x;
  const int half = lane >> 4;
  const int col = lane & 15;
  const int b = blockIdx.x * 16 + col;

  float cw[4], sw[4];
#pragma unroll
  for (int w = 0; w < 4; w++) {
    const float a = feat[b * 4 + w] * 0.5f;
    cw[w] = cosf(a);
    sw[w] = sinf(a);
  }

  v8f aR = {};
  v8f aI = {};
#pragma unroll
  for (int c = 0; c < 4; c++) {
    const int k0 = 4 * c + 2 * half;
    const int k1 = k0 + 1;
    // psi[k][col] for this lane's two K slots
    const float p0 = ((k0 & 8) ? sw[0] : cw[0]) * ((k0 & 4) ? sw[1] : cw[1]) *
                     ((k0 & 2) ? sw[2] : cw[2]) * ((k0 & 1) ? sw[3] : cw[3]);
    const float p1 = ((k1 & 8) ? sw[0] : cw[0]) * ((k1 & 4) ? sw[1] : cw[1]) *
                     ((k1 & 2) ? sw[2] : cw[2]) * ((k1 & 1) ? sw[3] : cw[3]);
    v2f B;
    B.x = p0;
    B.y = p1;
    v2f Ar;
    Ar.x = Ur[col * 16 + k0];
    Ar.y = Ur[col * 16 + k1];
    aR = __builtin_amdgcn_wmma_f32_16x16x4_f32(false, Ar, false, B, (short)0,
                                               aR, false, false);
    v2f Ai;
    Ai.x = Ui[col * 16 + k0];
    Ai.y = Ui[col * 16 + k1];
    aI = __builtin_amdgcn_wmma_f32_16x16x4_f32(false, Ai, false, B, (short)0,
                                               aI, false, false);
  }

  float e0 = 0.f, e1 = 0.f, e2 = 0.f, e3 = 0.f;
#pragma unroll
  for (int v = 0; v < 8; v++) {
    const int s = v + 8 * half;  // state index of this D element
    const float p = aR[v] * aR[v] + aI[v] * aI[v];
    e0 += (s & 8) ? -p : p;
    e1 += (s & 4) ? -p : p;
    e2 += (s & 2) ? -p : p;
    e3 += (s & 1) ? -p : p;
  }
  // combine the two lane-halves (same batch column)
  e0 += __shfl_xor(e0, 16, 32);
  e1 += __shfl_xor(e1, 16, 32);
  e2 += __shfl_xor(e2, 16, 32);
  e3 += __shfl_xor(e3, 16, 32);
  if (half == 0) {
    raw[b * 4 + 0] = e0;
    raw[b * 4 + 1] = e1;
    raw[b * 4 + 2] = e2;
    raw[b * 4 + 3] = e3;
  }
}

// ---------------------------------------------------------------------------
// Stage 4: BatchNorm1d (training mode, biased variance) over the 512 batch.
// ---------------------------------------------------------------------------
__global__ __launch_bounds__(512) void bn_kernel(const float* __restrict__ raw,
                                                 const float* __restrict__ gamma,
                                                 const float* __restrict__ beta,
                                                 float* __restrict__ out) {
  __shared__ float ssum[4], ssq[4];
  const int t = threadIdx.x;
  if (t < 4) { ssum[t] = 0.f; ssq[t] = 0.f; }
  __syncthreads();
  const float v0 = raw[t * 4 + 0], v1 = raw[t * 4 + 1];
  const float v2 = raw[t * 4 + 2], v3 = raw[t * 4 + 3];
  atomicAdd(&ssum[0], v0); atomicAdd(&ssq[0], v0 * v0);
  atomicAdd(&ssum[1], v1); atomicAdd(&ssq[1], v1 * v1);
  atomicAdd(&ssum[2], v2); atomicAdd(&ssq[2], v2 * v2);
  atomicAdd(&ssum[3], v3); atomicAdd(&ssq[3], v3 * v3);
  __syncthreads();
  const float vv[4] = {v0, v1, v2, v3};
#pragma unroll
  for (int w = 0; w < 4; w++) {
    const float m = ssum[w] * (1.0f / 512.0f);
    const float var = ssq[w] * (1.0f / 512.0f) - m * m;
    out[t * 4 + w] = gamma[w] * (vv[w] - m) * rsqrtf(var + 1e-5f) + beta[w];
  }
}

extern "C" void kernel_launch(void* const* d_in, const int* in_sizes, int n_in,
                              void* d_out, int out_size, void* d_ws,
                              size_t ws_size, hipStream_t stream) {
  const float* x = (const float*)d_in[0];       // (512,1,224,224)
  const float* conv_w = (const float*)d_in[1];  // (4,1,3,3)
  const float* params = (const float*)d_in[2];  // (3,4,2)
  const float* gamma = (const float*)d_in[3];   // (4,)
  const float* beta = (const float*)d_in[4];    // (4,)
  float* out = (float*)d_out;                   // (512,4)

  float* ws = (float*)d_ws;
  float* feat = ws;               // 512*4
  float* Ur = ws + 2048;          // 16*16
  float* Ui = ws + 2048 + 256;    // 16*16
  float* raw = ws + 2048 + 512;   // 512*4

  conv_pool_kernel<<<512, 224, 0, stream>>>(x, conv_w, feat);
  build_unitary_kernel<<<1, 16, 0, stream>>>(params, Ur, Ui);
  qexp_wmma_kernel<<<32, 32, 0, stream>>>(feat, Ur, Ui, raw);
  bn_kernel<<<1, 512, 0, stream>>>(raw, gamma, beta, out);
}